// SCRFDTDMMPostModel_14602888807051
// MI455X (gfx1250) — compile-verified
//
#include <hip/hip_runtime.h>
#include <hip/hip_bf16.h>
#include <math.h>

// ---------------------------------------------------------------------------
// SCRFD post-process for MI455X (gfx1250).
// Heavy op: (32000 x 228) x (228 x 204) f32 GEMM -> V_WMMA_F32_16X16X4_F32.
// f32 WMMA chosen deliberately: workload is HBM-bound (~31MB reads, ~1.3us at
// 23.3 TB/s) while the GEMM is only ~3 GFLOP, and the reference later does
// bit-exact box equality matching, so no precision can be given away.
// The B operand (204x228 basis, 186KB) is pre-packed once into the per-lane
// WMMA fragment layout so the GEMM inner loop is branch-free:
//   ds_load_b64 (A) + global_load_b64 (B, L2-resident) + v_wmma.
// ---------------------------------------------------------------------------

typedef __attribute__((ext_vector_type(2))) float v2f;
typedef __attribute__((ext_vector_type(8))) float v8f;

#define N_R_      9
#define PDIM_     237
#define KDIM_     228            // 199 shp + 29 exp
#define NDIM_     204
#define KSTEPS_   57             // 228 / 4
#define NTILES_   13             // ceil(204/16)
#define NB_       32
#define A16_      800
#define A32_      200
#define ROWS16_   (NB_*A16_)     // 25600
#define ROWS32_   (NB_*A32_)     // 6400
#define ROWSTOT_  (ROWS16_+ROWS32_)
#define CAND16_   (ROWS16_*2)    // 51200
#define CANDTOT_  (ROWSTOT_*2)   // 64000
#define NOBJ_     15
#define SENT_     (-1e30f)
#define KPTH_     0.95f
#define IOUTH_    0.45f

// workspace layout (4-byte element offsets; total ~5.11M elems = 20.5 MB)
static constexpr size_t OFF_LN      = 0;                                   // 32000*136 landmark (y,x) per anchor-row
static constexpr size_t OFF_CANDBOX = OFF_LN + (size_t)ROWSTOT_*136;       // 64000*4 candidate boxes (SENTINEL when unmasked)
static constexpr size_t OFF_PROB    = OFF_CANDBOX + (size_t)CANDTOT_*4;    // 64000
static constexpr size_t OFF_OUTB    = OFF_PROB + (size_t)CANDTOT_;         // 32*15*2*5 out_boxes (init -1)
static constexpr size_t OFF_SELBOX  = OFF_OUTB + (size_t)NB_*NOBJ_*2*5;    // 480*4 post-NMS boxes (-1 -> inf applied)
static constexpr size_t OFF_SELS    = OFF_SELBOX + (size_t)NB_*NOBJ_*4;    // 480
static constexpr size_t OFF_SELC    = OFF_SELS + (size_t)NB_*NOBJ_;        // 480
static constexpr size_t OFF_OUTLN   = OFF_SELC + (size_t)NB_*NOBJ_;        // 32*15*2*136 out landmarks (init -1)
static constexpr size_t OFF_MASKI   = OFF_OUTLN + (size_t)NB_*NOBJ_*2*136; // 64000 ints (view ws as int*)
static constexpr size_t OFF_IDXSI   = OFF_MASKI + (size_t)CANDTOT_;        // 64000*3 ints (b, oidx, c)
static constexpr size_t OFF_WPACK   = OFF_IDXSI + (size_t)CANDTOT_*3;      // 13*57*32*2 packed B fragments (8B aligned)

// B-matrix element W[n][k]: column n of combined (shp|exp) basis
__device__ __forceinline__ float wfetch(const float* __restrict__ shp,
                                        const float* __restrict__ expb,
                                        int j, int k) {
  if (j >= NDIM_) return 0.f;
  return (k < 199) ? shp[(size_t)j*199 + k] : expb[(size_t)j*29 + (k-199)];
}

// ---------------------------------------------------------------------------
// One-shot pack of the B operand into per-lane WMMA fragment order:
//   Wpack[((ntile*57 + kstep)*32 + lane)] = { W[n][k], W[n][k+1] }
//   n = ntile*16 + (lane&15), k = kstep*4 + ((lane>>4)<<1)
// Absorbs the shp/exp segment split and the n>=204 zero-pad.
// ---------------------------------------------------------------------------
__global__ void pack_b_kernel(const float* __restrict__ shp,
                              const float* __restrict__ expb,
                              float* __restrict__ ws) {
  int t = blockIdx.x*256 + threadIdx.x;          // 13*57*32 = 23712 fragments
  if (t >= NTILES_*KSTEPS_*32) return;
  int lane  = t & 31;
  int step  = (t >> 5) % KSTEPS_;
  int ntile = (t >> 5) / KSTEPS_;
  int n = ntile*16 + (lane & 15);
  int k = step*4 + ((lane >> 4) << 1);
  (ws + OFF_WPACK)[(size_t)t*2 + 0] = wfetch(shp, expb, n, k);
  (ws + OFF_WPACK)[(size_t)t*2 + 1] = wfetch(shp, expb, n, k + 1);
}

// ---------------------------------------------------------------------------
// Per-level fused kernel: coef staging -> f32 WMMA GEMM -> rotation ->
// min/max -> box scale -> candidate emit. 416 threads = 13 waves; wave w owns
// N-tile w (13*16 = 208 >= 204). grid.x = rows/16.
// ---------------------------------------------------------------------------
__global__ __launch_bounds__(416) void level_kernel(
    const float* __restrict__ cls, const float* __restrict__ bbox,
    const float* __restrict__ param, const float* __restrict__ pms,
    const float* __restrict__ u_base, const float* __restrict__ origin,
    float* __restrict__ ws, int rows, int A, int stride,
    int anchorRowOff, int candOff)
{
  __shared__ __align__(16) float coef_s[16][232]; // 16 x 228 coefficients (A tile)
  __shared__ float R_s[16][9];                    // per-row rotation
  __shared__ float ln_s[16][208];                 // GEMM out + u
  __shared__ float rot_s[16][68][2];              // rotated (x,y)
  __shared__ float scale_s[16][2];
  __shared__ float box_s[16][4];                  // x1,y1,x2,y2

  const int tid = threadIdx.x;
  const int rowBase = blockIdx.x * 16;
  const float* pms0 = pms;
  const float* pms1 = pms + PDIM_;

  for (int idx = tid; idx < 16*KDIM_; idx += 416) {
    int m = idx / KDIM_, k = idx - m*KDIM_;
    int row = rowBase + m;
    float v = 0.f;
    if (row < rows) { int pi = N_R_ + k; v = param[(size_t)row*PDIM_ + pi]*pms1[pi] + pms0[pi]; }
    coef_s[m][k] = v;
  }
  for (int idx = tid; idx < 16*9; idx += 416) {
    int m = idx / 9, k = idx - m*9;
    int row = rowBase + m;
    R_s[m][k] = (row < rows) ? (param[(size_t)row*PDIM_ + k]*pms1[k] + pms0[k]) : 0.f;
  }
  __syncthreads();

  // ---- GEMM: D(16x16) += A(16x4) x B(4x16), 57 K-steps, branch-free -------
  const int wave  = tid >> 5;
  const int lane  = tid & 31;
  const int mloc  = lane & 15;           // M for A frag
  const int khalf = (lane >> 4) << 1;    // lanes 16-31 hold K+2,K+3
  const v2f* __restrict__ wp =
      (const v2f*)(ws + OFF_WPACK) + ((size_t)wave*KSTEPS_)*32 + lane;
  v8f acc = {};
  #pragma unroll
  for (int t = 0; t < KSTEPS_; ++t) {
    v2f a = *(const v2f*)&coef_s[mloc][t*4 + khalf];   // ds_load_b64
    v2f b = wp[(size_t)t*32];                          // global_load_b64 (L2-hit)
    acc = __builtin_amdgcn_wmma_f32_16x16x4_f32(
        /*neg_a=*/false, a, /*neg_b=*/false, b,
        /*c_mod=*/(short)0, acc, /*reuse_a=*/false, /*reuse_b=*/false);
  }
  const int ncol = wave*16 + mloc;
  if (ncol < NDIM_) {
    float u = u_base[ncol];
    const int mhi = (lane >> 4) << 3;    // C layout: VGPR v -> M=v (+8 for hi lanes)
    #pragma unroll
    for (int vv = 0; vv < 8; ++vv) ln_s[mhi + vv][ncol] = acc[vv] + u;
  }
  __syncthreads();

  // ---- rotate 68 points by R[0:2,:] ---------------------------------------
  for (int t = tid; t < 16*68; t += 416) {
    int m = t / 68, kp = t - m*68;
    float x = ln_s[m][kp*3+0], y = ln_s[m][kp*3+1], z = ln_s[m][kp*3+2];
    rot_s[m][kp][0] = x*R_s[m][0] + y*R_s[m][1] + z*R_s[m][2];
    rot_s[m][kp][1] = x*R_s[m][3] + y*R_s[m][4] + z*R_s[m][5];
  }
  __syncthreads();

  // ---- per (row, component): min/max + box + scale ------------------------
  if (tid < 32) {
    int m = tid >> 1, comp = tid & 1;
    float tl = rot_s[m][0][comp], br = tl;
    for (int kp = 1; kp < 68; ++kp) {
      float v = rot_s[m][kp][comp];
      tl = fminf(tl, v); br = fmaxf(br, v);
    }
    int row = rowBase + m;
    float sc = 0.f;
    if (row < rows) {
      int a = row % A;
      int loc = a >> 1;                        // NUM_ANCHORS = 2
      int hw = 320 / stride;
      float acx = (float)((loc % hw) * stride);
      float acy = (float)((loc / hw) * stride);
      float d0 = bbox[(size_t)row*4+0]*(float)stride;
      float d1 = bbox[(size_t)row*4+1]*(float)stride;
      float d2 = bbox[(size_t)row*4+2]*(float)stride;
      float d3 = bbox[(size_t)row*4+3]*(float)stride;
      float x1 = acx - d0, y1 = acy - d1, x2 = acx + d2, y2 = acy + d3;
      float lo = comp ? y1 : x1, hi = comp ? y2 : x2;
      sc = fabsf((hi - lo) / (br - tl));
      if (comp == 0) { box_s[m][0]=x1; box_s[m][1]=y1; box_s[m][2]=x2; box_s[m][3]=y2; }
    }
    scale_s[m][comp] = sc;
  }
  __syncthreads();

  const float r0 = origin[0] * (1.f/320.f);
  const float r1 = origin[1] * (1.f/320.f);

  // ---- emit landmarks (y,x)*r ---------------------------------------------
  float* lnArr = ws + OFF_LN;
  for (int t = tid; t < 16*68; t += 416) {
    int m = t / 68, kp = t - m*68;
    int row = rowBase + m;
    if (row < rows) {
      float X = scale_s[m][0] * rot_s[m][kp][0];
      float Y = scale_s[m][1] * rot_s[m][kp][1];
      size_t g = (size_t)(anchorRowOff + row) * 136 + (size_t)kp*2;
      lnArr[g]   = Y * r0;
      lnArr[g+1] = X * r1;
    }
  }

  // ---- emit per-(row,class) candidates ------------------------------------
  if (tid < 32) {
    int m = tid >> 1, c = tid & 1;
    int row = rowBase + m;
    if (row < rows) {
      float logit = cls[(size_t)row*2 + c];
      float prob = 1.f / (1.f + __expf(-logit));
      int msk = prob > KPTH_;
      float yx0 = box_s[m][1]*r0, yx1 = box_s[m][0]*r1;
      float yx2 = box_s[m][3]*r0, yx3 = box_s[m][2]*r1;
      size_t gc = (size_t)candOff + (size_t)row*2 + c;
      float* cb = ws + OFF_CANDBOX + gc*4;
      cb[0] = msk ? yx0 : SENT_;  cb[1] = msk ? yx1 : SENT_;
      cb[2] = msk ? yx2 : SENT_;  cb[3] = msk ? yx3 : SENT_;
      (ws + OFF_PROB)[gc] = prob;
      ((int*)ws)[OFF_MASKI + gc] = msk;
    }
  }
}

// ---------------------------------------------------------------------------
__global__ void init_kernel(float* __restrict__ ws) {
  size_t i = (size_t)blockIdx.x*256 + threadIdx.x;
  if (i < (size_t)NB_*NOBJ_*2*5)   (ws + OFF_OUTB)[i]  = -1.f;
  if (i < (size_t)NB_*NOBJ_*2*136) (ws + OFF_OUTLN)[i] = -1.f;
}

// ---------------------------------------------------------------------------
// Single-block global prefix sum per level (replicates the reference's
// cumsum over the flattened (B,A,C) mask) + drop-scatter into out_boxes.
// ---------------------------------------------------------------------------
__global__ __launch_bounds__(1024) void scan_scatter_kernel(float* __restrict__ ws) {
  __shared__ int sc[1024];
  const int tid = threadIdx.x;
  const int* maskArr = ((const int*)ws) + OFF_MASKI;
  int* idxs = ((int*)ws) + OFF_IDXSI;
  const float* candBox = ws + OFF_CANDBOX;
  const float* probArr = ws + OFF_PROB;
  float* outB = ws + OFF_OUTB;

  const int segStart[2] = {0, CAND16_};
  const int segEnd[2]   = {CAND16_, CANDTOT_};
  const int Alev[2]     = {A16_, A32_};

  for (int L = 0; L < 2; ++L) {
    int running = 0;
    for (int base = segStart[L]; base < segEnd[L]; base += 1024) {
      int i = base + tid;
      int m = (i < segEnd[L]) ? maskArr[i] : 0;
      sc[tid] = m;
      __syncthreads();
      for (int off = 1; off < 1024; off <<= 1) {
        int t = (tid >= off) ? sc[tid - off] : 0;
        __syncthreads();
        sc[tid] += t;
        __syncthreads();
      }
      int incl  = sc[tid];
      int total = sc[1023];
      if (i < segEnd[L]) {
        int oidx = running + incl - 1;
        int e = i - segStart[L];
        int rowLevel = e >> 1;
        int b = rowLevel / Alev[L];
        int c = e & 1;
        idxs[(size_t)i*3+0] = b;
        idxs[(size_t)i*3+1] = m ? oidx : NOBJ_;
        idxs[(size_t)i*3+2] = c;
        if (m && oidx < NOBJ_) {
          float* o = outB + (((size_t)b*NOBJ_ + oidx)*2 + c)*5;
          o[0] = candBox[(size_t)i*4+0];
          o[1] = candBox[(size_t)i*4+1];
          o[2] = candBox[(size_t)i*4+2];
          o[3] = candBox[(size_t)i*4+3];
          o[4] = probArr[i];
        }
      }
      running += total;
      __syncthreads();
    }
  }
}

// ---------------------------------------------------------------------------
// Per-batch greedy NMS over 15x2 slots + stable top-15 merge. Tiny -> serial.
// ---------------------------------------------------------------------------
__global__ void nms_kernel(float* __restrict__ ws) {
  if (threadIdx.x != 0) return;
  const int b = blockIdx.x;
  const float* outB = ws + OFF_OUTB + (size_t)b*NOBJ_*2*5;

  float cb[30][4], cs_[30], cc[30];
  for (int c = 0; c < 2; ++c) {
    float bx[15][4], s[15], area[15];
    for (int i = 0; i < 15; ++i) {
      const float* o = outB + ((size_t)i*2 + c)*5;
      bx[i][0]=o[0]; bx[i][1]=o[1]; bx[i][2]=o[2]; bx[i][3]=o[3]; s[i]=o[4];
      area[i] = fmaxf(bx[i][2]-bx[i][0],0.f) * fmaxf(bx[i][3]-bx[i][1],0.f);
    }
    bool act[15]; int sel[15];
    for (int i = 0; i < 15; ++i) act[i] = true;
    for (int it = 0; it < 15; ++it) {
      float best = -INFINITY; int j = 0;
      for (int i = 0; i < 15; ++i) {
        float v = act[i] ? s[i] : -INFINITY;
        if (v > best) { best = v; j = i; }
      }
      bool valid = best > -INFINITY;
      sel[it] = valid ? j : -1;
      for (int i = 0; i < 15; ++i) {
        float t0 = fmaxf(bx[j][0], bx[i][0]), t1 = fmaxf(bx[j][1], bx[i][1]);
        float b0 = fminf(bx[j][2], bx[i][2]), b1 = fminf(bx[j][3], bx[i][3]);
        float inter = fmaxf(b0-t0, 0.f) * fmaxf(b1-t1, 0.f);
        float un = area[j] + area[i] - inter;
        float iou = (un > 0.f) ? inter/un : 0.f;
        act[i] = act[i] && (iou <= IOUTH_) && valid;
      }
      act[j] = false;
    }
    for (int i = 0; i < 15; ++i) {
      int si = sel[i] < 0 ? 0 : sel[i];
      bool valid = sel[i] >= 0;
      int k = c*15 + i;
      cb[k][0]=bx[si][0]; cb[k][1]=bx[si][1]; cb[k][2]=bx[si][2]; cb[k][3]=bx[si][3];
      cs_[k] = valid ? s[si] : -INFINITY;
      cc[k]  = (float)c;
    }
  }
  // stable descending top-15 (strict > keeps earliest index; -inf group stable)
  bool used[30]; for (int i = 0; i < 30; ++i) used[i] = false;
  for (int slot = 0; slot < 15; ++slot) {
    float best = -INFINITY; int j = -1;
    for (int i = 0; i < 30; ++i)
      if (!used[i] && cs_[i] > best) { best = cs_[i]; j = i; }
    if (j < 0) for (int i = 0; i < 30; ++i) if (!used[i]) { j = i; break; }
    used[j] = true;
    float ss = cs_[j];
    bool valid = ss > -INFINITY;
    float* sb = ws + OFF_SELBOX + ((size_t)b*NOBJ_ + slot)*4;
    #pragma unroll
    for (int q = 0; q < 4; ++q) {
      float v = valid ? cb[j][q] : 0.f;       // invalid -> 0 (reference)
      sb[q] = (v == -1.f) ? INFINITY : v;     // then -1 -> inf
    }
    (ws + OFF_SELS)[(size_t)b*NOBJ_ + slot] = valid ? ss : 0.f;
    (ws + OFF_SELC)[(size_t)b*NOBJ_ + slot] = valid ? cc[j] : 0.f;
  }
}

// ---------------------------------------------------------------------------
// Exact-equality match of each NMS box against the 64000 candidates
// (first-match index == argmax semantics), then scatter landmarks.
// ---------------------------------------------------------------------------
__global__ __launch_bounds__(256) void match_kernel(float* __restrict__ ws) {
  __shared__ int smin[256];
  __shared__ int s_bc[4];        // idx0, idx1, idx2, cand
  const int i = blockIdx.x;      // 0..B*NOBJ-1
  const float* tgt = ws + OFF_SELBOX + (size_t)i*4;
  const float t0 = tgt[0], t1 = tgt[1], t2 = tgt[2], t3 = tgt[3];
  const float* candBox = ws + OFF_CANDBOX;

  int mn = 0x7fffffff;
  for (int j = threadIdx.x; j < CANDTOT_; j += 256) {
    const float* cbp = candBox + (size_t)j*4;
    if (cbp[0]==t0 && cbp[1]==t1 && cbp[2]==t2 && cbp[3]==t3) mn = min(mn, j);
  }
  smin[threadIdx.x] = mn;
  __syncthreads();
  for (int off = 128; off > 0; off >>= 1) {
    if (threadIdx.x < off) smin[threadIdx.x] = min(smin[threadIdx.x], smin[threadIdx.x+off]);
    __syncthreads();
  }
  if (threadIdx.x == 0) {
    int cand = smin[0];
    bool matched = cand != 0x7fffffff;
    if (!matched) cand = 0;
    const int* idxs = ((const int*)ws) + OFF_IDXSI;
    if (matched) { s_bc[0]=idxs[(size_t)cand*3]; s_bc[1]=idxs[(size_t)cand*3+1]; s_bc[2]=idxs[(size_t)cand*3+2]; }
    else         { s_bc[0]=0; s_bc[1]=NOBJ_; s_bc[2]=0; }
    s_bc[3] = cand;
  }
  __syncthreads();
  const int oidx = s_bc[1];
  if (oidx < NOBJ_) {
    const int cand = s_bc[3];
    const int aRow = (cand < CAND16_) ? (cand >> 1) : (ROWS16_ + ((cand - CAND16_) >> 1));
    const float* lnSrc = ws + OFF_LN + (size_t)aRow*136;
    float* lnDst = ws + OFF_OUTLN + (((size_t)s_bc[0]*NOBJ_ + oidx)*2 + s_bc[2])*136;
    for (int k = threadIdx.x; k < 136; k += 256) lnDst[k] = lnSrc[k];
  }
}

// ---------------------------------------------------------------------------
__global__ void finalize_kernel(const float* __restrict__ ws, float* __restrict__ out) {
  const int NBB = NB_*NOBJ_*6;                 // 2880
  const int NLN = NB_*NOBJ_*2*136;             // 130560
  int i = blockIdx.x*256 + threadIdx.x;
  if (i < NBB) {
    int r = i / 6, comp = i - r*6;
    float v;
    if (comp < 4) {
      v = (ws + OFF_SELBOX)[(size_t)r*4 + comp];
      if (v - 1.f == -1.f) v = INFINITY;       // box == 0 -> inf (reference quirk)
    } else if (comp == 4) {
      v = (ws + OFF_SELS)[r];
    } else {
      v = (ws + OFF_SELC)[r];
    }
    if (v == -1.f) v = INFINITY;
    out[i] = v;
  } else if (i < NBB + NLN) {
    float v = (ws + OFF_OUTLN)[i - NBB];
    if (v == -1.f) v = INFINITY;
    out[i] = v;
  }
}

// ---------------------------------------------------------------------------
extern "C" void kernel_launch(void* const* d_in, const int* in_sizes, int n_in,
                              void* d_out, int out_size, void* d_ws, size_t ws_size,
                              hipStream_t stream) {
  // setup_inputs order: cls_8, bbox_8, param_8, cls_16, bbox_16, param_16,
  //                     cls_32, bbox_32, param_32, origin_shapes, pms,
  //                     u_base, shp_base, exp_base.  (_8 level unused.)
  const float* cls16    = (const float*)d_in[3];
  const float* bbox16   = (const float*)d_in[4];
  const float* param16  = (const float*)d_in[5];
  const float* cls32    = (const float*)d_in[6];
  const float* bbox32   = (const float*)d_in[7];
  const float* param32  = (const float*)d_in[8];
  const float* origin   = (const float*)d_in[9];
  const float* pms      = (const float*)d_in[10];
  const float* u_base   = (const float*)d_in[11];
  const float* shp_base = (const float*)d_in[12];
  const float* exp_base = (const float*)d_in[13];
  float* ws  = (float*)d_ws;
  float* out = (float*)d_out;

  init_kernel<<<(NB_*NOBJ_*2*136 + 255)/256, 256, 0, stream>>>(ws);
  pack_b_kernel<<<(NTILES_*KSTEPS_*32 + 255)/256, 256, 0, stream>>>(shp_base, exp_base, ws);

  level_kernel<<<ROWS16_/16, 416, 0, stream>>>(
      cls16, bbox16, param16, pms, u_base, origin,
      ws, ROWS16_, A16_, 16, /*anchorRowOff=*/0, /*candOff=*/0);

  level_kernel<<<ROWS32_/16, 416, 0, stream>>>(
      cls32, bbox32, param32, pms, u_base, origin,
      ws, ROWS32_, A32_, 32, /*anchorRowOff=*/ROWS16_, /*candOff=*/CAND16_);

  scan_scatter_kernel<<<1, 1024, 0, stream>>>(ws);
  nms_kernel<<<NB_, 32, 0, stream>>>(ws);
  match_kernel<<<NB_*NOBJ_, 256, 0, stream>>>(ws);
  finalize_kernel<<<(NB_*NOBJ_*6 + NB_*NOBJ_*2*136 + 255)/256, 256, 0, stream>>>(ws, out);
}